// BinaryHashGrid_69372311765526
// MI455X (gfx1250) — compile-verified
//
#include <hip/hip_runtime.h>
#include <stdint.h>

typedef uint32_t u32;

// ---- problem constants -----------------------------------------------------
#define NL3 12
#define NL2 4
#define T3N 8192u      // 2^13
#define T2N 32768u     // 2^15
#define T3_MASK 8191u
#define T2_MASK 32767u
#define PRIME1 2654435761u
#define PRIME2 805459861u

// packed-table layout inside d_ws / LDS (in dwords; 8 entries(4-bit nibbles)/dword)
//   3D: level l at dword l*1024              (12 * 1024  = 12288 dwords)
//   2D: plane p (0=xy,1=xz,2=yz), level l at 12288 + (p*4+l)*4096
#define DW_3D_PER_LVL 1024
#define DW_2D_PER_TBL 4096
#define DW_TOTAL 61440          // 12288 + 12*4096
#define LDS_BYTES (DW_TOTAL * 4)  // 245760 B  (fits CDNA5 320KB/WGP LDS)

__device__ __forceinline__ u32 f2u(float x) { return __float_as_uint(x); }
__device__ __forceinline__ float u2f(u32 x) { return __uint_as_float(x); }

// ---- kernel 1: pack sign bits (4 features -> 1 nibble, 8 entries -> 1 dword)
__global__ __launch_bounds__(256) void bhg_pack_signs(
    const float* __restrict__ t3, const float* __restrict__ txy,
    const float* __restrict__ txz, const float* __restrict__ tyz,
    u32* __restrict__ ws) {
  int i = blockIdx.x * blockDim.x + threadIdx.x;   // dword index
  if (i >= DW_TOTAL) return;
  const float* src;
  if (i < NL3 * DW_3D_PER_LVL) {
    int level = i >> 10;
    int entry = (i & 1023) << 3;
    src = t3 + ((size_t)level * T3N + entry) * 4;
  } else {
    int j = i - NL3 * DW_3D_PER_LVL;
    int tb = j >> 12;                 // 0..11 = plane*4 + level
    int entry = (j & 4095) << 3;
    const float* base = (tb < 4) ? txy : (tb < 8) ? txz : tyz;
    int level = tb & 3;
    src = base + ((size_t)level * T2N + entry) * 4;
  }
  u32 v = 0;
#pragma unroll
  for (int e = 0; e < 8; ++e)
#pragma unroll
    for (int f = 0; f < 4; ++f)
      v |= (src[e * 4 + f] < 0.0f ? 1u : 0u) << (e * 4 + f);  // bit=1 -> -1
  ws[i] = v;
}

// branch-free signed accumulate: a_f += (nib bit f ? -w : +w)
__device__ __forceinline__ void acc4(u32 nib, float w,
                                     float& a0, float& a1, float& a2, float& a3) {
  u32 wu = f2u(w);
  a0 += u2f(wu ^ ((nib << 31) & 0x80000000u));
  a1 += u2f(wu ^ ((nib << 30) & 0x80000000u));
  a2 += u2f(wu ^ ((nib << 29) & 0x80000000u));
  a3 += u2f(wu ^ ((nib << 28) & 0x80000000u));
}

// ---- per-plane bilinear accumulate from packed LDS table -------------------
__device__ __forceinline__ void plane2d(const u32* __restrict__ tb,
                                        float u, float v, float r,
                                        float& a0, float& a1, float& a2, float& a3) {
  float su = u * r, sv = v * r;
  float fu = floorf(su), fv = floorf(sv);
  u32 iu = (u32)fu, iv = (u32)fv;
  float wu = su - fu, wv = sv - fv;
  u32 hu[2] = {iu, iu + 1u};
  u32 hv[2] = {iv * PRIME1, (iv + 1u) * PRIME1};
  float vu[2] = {1.0f - wu, wu};
  float vv[2] = {1.0f - wv, wv};
#pragma unroll
  for (int du = 0; du < 2; ++du)
#pragma unroll
    for (int dv = 0; dv < 2; ++dv) {
      u32 idx = (hu[du] ^ hv[dv]) & T2_MASK;
      u32 nib = (tb[idx >> 3] >> ((idx & 7u) << 2)) & 0xFu;
      acc4(nib, vu[du] * vv[dv], a0, a1, a2, a3);
    }
}

// ---- kernel 2: main encode -------------------------------------------------
__global__ __launch_bounds__(1024) void bhg_main(
    const float* __restrict__ pos, const float* __restrict__ bmin,
    const float* __restrict__ bmax, const u32* __restrict__ ws,
    float* __restrict__ out, int npts) {
  extern __shared__ u32 lds[];

  // ---- stage packed tables into LDS ----
#if defined(__HIP_DEVICE_COMPILE__) && defined(__gfx1250__) && \
    __has_builtin(__builtin_amdgcn_tensor_load_to_lds)
  // CDNA5 Tensor Data Mover: one 1-D DMA of 61440 dwords into LDS offset 0.
  if (threadIdx.x == 0) {
    typedef u32 u32x4 __attribute__((ext_vector_type(4)));
    typedef int i32x8 __attribute__((ext_vector_type(8)));
    typedef int i32x4 __attribute__((ext_vector_type(4)));
    uint64_t ga = (uint64_t)(uintptr_t)ws;
    u32x4 g0;
    g0[0] = 1u;                                    // count=1, user mode
    g0[1] = 0u;                                    // lds_addr = 0 (no static LDS)
    g0[2] = (u32)(ga & 0xffffffffu);               // global_addr[31:0]
    g0[3] = ((u32)(ga >> 32) & 0x01ffffffu) | (2u << 30);  // addr[56:32], type=2
    i32x8 g1;
    g1[0] = (int)(2u << 16);                       // data_size=2 -> 4 bytes
    g1[1] = (int)((DW_TOTAL & 0xffffu) << 16);     // tensor_dim0[15:0]
    g1[2] = (int)(((DW_TOTAL >> 16) & 0xffffu) | (1u << 16)); // dim0 hi | tensor_dim1=1
    g1[3] = (int)((DW_TOTAL & 0xffffu) << 16);     // tile_dim0 = 61440
    g1[4] = 0;                                     // tile_dim1/2 unused
    g1[5] = (int)DW_TOTAL;                         // tensor_dim0_stride
    g1[6] = 0;
    g1[7] = 0;
    i32x4 z4 = {0, 0, 0, 0};
    i32x8 z8 = {0, 0, 0, 0, 0, 0, 0, 0};
    __builtin_amdgcn_tensor_load_to_lds(g0, g1, z4, z4, z8, 0);  // 6-arg (clang-23)
    __builtin_amdgcn_s_wait_tensorcnt(0);
  }
  __syncthreads();
#else
  // fallback: cooperative vectorized copy
  {
    const uint4* src = (const uint4*)ws;
    uint4* dst = (uint4*)lds;
    for (u32 i = threadIdx.x; i < DW_TOTAL / 4; i += blockDim.x) dst[i] = src[i];
    __syncthreads();
  }
#endif

  const float res3[NL3] = {16.f, 21.f, 30.f, 41.f, 56.f, 77.f,
                           105.f, 145.f, 198.f, 272.f, 373.f, 512.f};
  const float res2[NL2] = {128.f, 256.f, 511.f, 1024.f};  // 8^(2/3) double rounds < 4

  const float mnx = bmin[0], mny = bmin[1], mnz = bmin[2];
  const float isx = 1.0f / (bmax[0] - mnx + 1e-8f);
  const float isy = 1.0f / (bmax[1] - mny + 1e-8f);
  const float isz = 1.0f / (bmax[2] - mnz + 1e-8f);

  int n = blockIdx.x * 1024 + (int)threadIdx.x;   // 1 point per thread
  if (n >= npts) return;
  float px = pos[3 * n + 0], py = pos[3 * n + 1], pz = pos[3 * n + 2];
  float nx = fminf(fmaxf((px - mnx) * isx, 0.0f), 0.9999f);
  float ny = fminf(fmaxf((py - mny) * isy, 0.0f), 0.9999f);
  float nz = fminf(fmaxf((pz - mnz) * isz, 0.0f), 0.9999f);
  float* op = out + (size_t)n * 64;

  // ---- 12 3D levels ----
#pragma unroll
  for (int l = 0; l < NL3; ++l) {
    float r = res3[l];
    float sx = nx * r, sy = ny * r, sz = nz * r;
    float fx = floorf(sx), fy = floorf(sy), fz = floorf(sz);
    u32 ix = (u32)fx, iy = (u32)fy, iz = (u32)fz;
    float wx = sx - fx, wy = sy - fy, wz = sz - fz;
    u32 hx[2] = {ix, ix + 1u};
    u32 hy[2] = {iy * PRIME1, (iy + 1u) * PRIME1};
    u32 hz[2] = {iz * PRIME2, (iz + 1u) * PRIME2};
    float vx[2] = {1.0f - wx, wx};
    float vy[2] = {1.0f - wy, wy};
    float vz[2] = {1.0f - wz, wz};
    const u32* tb = lds + l * DW_3D_PER_LVL;
    float a0 = 0.f, a1 = 0.f, a2 = 0.f, a3 = 0.f;
#pragma unroll
    for (int dx = 0; dx < 2; ++dx)
#pragma unroll
      for (int dy = 0; dy < 2; ++dy) {
        u32 hxy = hx[dx] ^ hy[dy];
        float wxy = vx[dx] * vy[dy];
#pragma unroll
        for (int dz = 0; dz < 2; ++dz) {
          u32 idx = (hxy ^ hz[dz]) & T3_MASK;
          u32 nib = (tb[idx >> 3] >> ((idx & 7u) << 2)) & 0xFu;
          acc4(nib, wxy * vz[dz], a0, a1, a2, a3);
        }
      }
    float4 st = make_float4(a0, a1, a2, a3);
    *(float4*)(op + l * 4) = st;
  }

  // ---- 4 2D levels (xy + xz + yz summed) ----
#pragma unroll
  for (int l = 0; l < NL2; ++l) {
    float r = res2[l];
    float a0 = 0.f, a1 = 0.f, a2 = 0.f, a3 = 0.f;
    const u32* base = lds + NL3 * DW_3D_PER_LVL;
    plane2d(base + (0 * 4 + l) * DW_2D_PER_TBL, nx, ny, r, a0, a1, a2, a3);
    plane2d(base + (1 * 4 + l) * DW_2D_PER_TBL, nx, nz, r, a0, a1, a2, a3);
    plane2d(base + (2 * 4 + l) * DW_2D_PER_TBL, ny, nz, r, a0, a1, a2, a3);
    float4 st = make_float4(a0, a1, a2, a3);
    *(float4*)(op + 48 + l * 4) = st;
  }
}

// ---- host glue -------------------------------------------------------------
extern "C" void kernel_launch(void* const* d_in, const int* in_sizes, int n_in,
                              void* d_out, int out_size, void* d_ws, size_t ws_size,
                              hipStream_t stream) {
  const float* pos  = (const float*)d_in[0];
  const float* t3   = (const float*)d_in[1];
  const float* txy  = (const float*)d_in[2];
  const float* txz  = (const float*)d_in[3];
  const float* tyz  = (const float*)d_in[4];
  const float* bmin = (const float*)d_in[5];
  const float* bmax = (const float*)d_in[6];
  u32* ws = (u32*)d_ws;
  float* out = (float*)d_out;
  int npts = in_sizes[0] / 3;

  // allow 240KB dynamic LDS (CDNA5 supports up to 320KB per workgroup)
  (void)hipFuncSetAttribute((const void*)bhg_main,
                            hipFuncAttributeMaxDynamicSharedMemorySize, LDS_BYTES);

  bhg_pack_signs<<<(DW_TOTAL + 255) / 256, 256, 0, stream>>>(t3, txy, txz, tyz, ws);
  int blocks = (npts + 1023) / 1024;
  bhg_main<<<blocks, 1024, LDS_BYTES, stream>>>(pos, bmin, bmax, ws, out, npts);
}